// SAGE_62646392979926
// MI455X (gfx1250) — compile-verified
//
#include <hip/hip_runtime.h>

// ---------------------------------------------------------------------------
// GraphSAGE (3-layer) for MI455X / gfx1250.
//   Phase A (per layer): edge scatter-add; agg (25.6MB) is L2-resident
//                        (192MB L2), bound by L2 fp32-atomic throughput.
//   Phase B (per layer): dense combine via V_WMMA_F32_16X16X4_F32 with both
//                        weight matrices staged in LDS (CDNA5: 320KB/WGP, so
//                        Wself+Wneigh = 128KB fit at once). Each block owns a
//                        128-row slab; 8 waves cover 8 row-tiles x col-tiles.
// ---------------------------------------------------------------------------

typedef float v2f __attribute__((ext_vector_type(2)));
typedef float v8f __attribute__((ext_vector_type(8)));

static __host__ inline int cdiv(int a, int b) { return (a + b - 1) / b; }

// ---- zero fill -------------------------------------------------------------
__global__ void fill0_f32(float* __restrict__ p, int n) {
    int i = blockIdx.x * blockDim.x + threadIdx.x;
    if (i < n) p[i] = 0.0f;
}

__global__ void fill0_f32x4(float4* __restrict__ p, int n4) {
    int i = blockIdx.x * blockDim.x + threadIdx.x;
    if (i < n4) p[i] = make_float4(0.f, 0.f, 0.f, 0.f);
}

// ---- in-degree -------------------------------------------------------------
__global__ void deg_count(const int* __restrict__ dst, float* __restrict__ deg, int E) {
    int e = blockIdx.x * blockDim.x + threadIdx.x;
    if (e < E) unsafeAtomicAdd(deg + dst[e], 1.0f);
}

__global__ void deg_invert(float* __restrict__ deg, int n) {
    int i = blockIdx.x * blockDim.x + threadIdx.x;
    if (i < n) deg[i] = 1.0f / fmaxf(deg[i], 1.0f);
}

// ---- edge aggregation: agg[dst] += h[src], d = 128 fixed -------------------
// 32 threads per edge; each thread owns one float4 feature chunk (coalesced
// gather from h, 4 scalar fp32 atomics into L2-resident agg).
__global__ void scatter_add_128(const float* __restrict__ h,
                                const int* __restrict__ src,
                                const int* __restrict__ dst,
                                float* __restrict__ agg, int E) {
    int t = blockIdx.x * blockDim.x + threadIdx.x;
    int e = t >> 5;
    int l = t & 31;
    if (e >= E) return;
    int s = src[e];
    int d = dst[e];
    const float4 v = *(const float4*)(h + (size_t)s * 128 + l * 4);
    float* a = agg + (size_t)d * 128 + l * 4;
    unsafeAtomicAdd(a + 0, v.x);
    unsafeAtomicAdd(a + 1, v.y);
    unsafeAtomicAdd(a + 2, v.z);
    unsafeAtomicAdd(a + 3, v.w);
}

// ---- fused SAGE combine: out = act(h@Wself + (agg*deg_inv)@Wneigh + b) -----
// Block = 256 threads (8 wave32), owns rows [blockIdx*128, +128).
// Both weight matrices staged in dynamic LDS once per block; B-fragments read
// via ds_load. One wave per 16x16 output tile, V_WMMA_F32_16X16X4_F32:
//   A (16x4, 2 VGPR): row = lane&15, k-halves by lane>>4
//   B (4x16, 2 VGPR): col = lane&15, same k-half split
//   C/D (16x16, 8 VGPR): row = v + (lane>>4)*8, col = lane&15
__global__ void sage_combine_wmma(const float* __restrict__ h,
                                  const float* __restrict__ agg,
                                  const float* __restrict__ deg_inv,
                                  const float* __restrict__ Wself,
                                  const float* __restrict__ Wneigh,
                                  const float* __restrict__ bias,
                                  float* __restrict__ out,
                                  int n, int din, int dout, int relu) {
    extern __shared__ float smem[];          // [din*dout] Wself | [din*dout] Wneigh
    float* sWs = smem;
    float* sWn = smem + (size_t)din * dout;

    // cooperative float4 stage of both weight matrices into LDS
    {
        const int    nw4 = (din * dout) >> 2;
        const float4* g0 = (const float4*)Wself;
        const float4* g1 = (const float4*)Wneigh;
        float4*       s0 = (float4*)sWs;
        float4*       s1 = (float4*)sWn;
        for (int i = threadIdx.x; i < nw4; i += blockDim.x) {
            s0[i] = g0[i];
            s1[i] = g1[i];
        }
    }
    __syncthreads();

    const int wave   = threadIdx.x >> 5;
    const int lane   = threadIdx.x & 31;
    const int tilesN = dout >> 4;
    const int nTiles = 8 * tilesN;           // 8 row-tiles x tilesN col-tiles
    const int rowBase = blockIdx.x * 128;

    const int lr   = lane & 15;
    const int koff = (lane >> 4) * 2;        // which K-pair this half-wave holds

    for (int t = wave; t < nTiles; t += 8) {
        int rt = t / tilesN;
        int tn = t - rt * tilesN;
        int r0 = rowBase + rt * 16;
        if (r0 + 16 > n) continue;           // wave-uniform: EXEC stays all-1s

        int row = r0 + lr;
        int col = tn * 16 + lr;
        const float dinv = deg_inv[row];
        const float* hrow = h   + (size_t)row * din;
        const float* arow = agg + (size_t)row * din;

        v8f c = {0.f, 0.f, 0.f, 0.f, 0.f, 0.f, 0.f, 0.f};

        for (int k = 0; k < din; k += 4) {
            v2f a0 = *(const v2f*)(hrow + k + koff);   // self A-frag
            v2f a1 = *(const v2f*)(arow + k + koff);   // neigh A-frag
            a1 *= dinv;                                // fold mean-divide
            v2f b0, b1;                                // B-frags from LDS
            b0.x = sWs[(k + koff) * dout + col];
            b0.y = sWs[(k + koff + 1) * dout + col];
            b1.x = sWn[(k + koff) * dout + col];
            b1.y = sWn[(k + koff + 1) * dout + col];
            c = __builtin_amdgcn_wmma_f32_16x16x4_f32(false, a0, false, b0,
                                                      (short)0, c, false, false);
            c = __builtin_amdgcn_wmma_f32_16x16x4_f32(false, a1, false, b1,
                                                      (short)0, c, false, false);
        }

        // writeback with bias (+ optional ReLU)
        int   crow0 = r0 + (lane >> 4) * 8;
        float bv    = bias[col];
        #pragma unroll
        for (int v = 0; v < 8; ++v) {
            float val = c[v] + bv;
            if (relu) val = fmaxf(val, 0.f);
            out[(size_t)(crow0 + v) * dout + col] = val;
        }
    }
}

// ---------------------------------------------------------------------------
extern "C" void kernel_launch(void* const* d_in, const int* in_sizes, int n_in,
                              void* d_out, int out_size, void* d_ws, size_t ws_size,
                              hipStream_t stream) {
    const float* x   = (const float*)d_in[0];
    const int*   src = (const int*)d_in[1];
    const int*   dst = (const int*)d_in[2];
    const float* Ws0 = (const float*)d_in[3];
    const float* Wn0 = (const float*)d_in[4];
    const float* b0  = (const float*)d_in[5];
    const float* Ws1 = (const float*)d_in[6];
    const float* Wn1 = (const float*)d_in[7];
    const float* b1  = (const float*)d_in[8];
    const float* Ws2 = (const float*)d_in[9];
    const float* Wn2 = (const float*)d_in[10];
    const float* b2  = (const float*)d_in[11];
    float*       out = (float*)d_out;

    const int N = in_sizes[0] / 128;   // 50000
    const int E = in_sizes[1];         // 600000
    const int HID = 128, OUT = 64;

    // workspace carve-up (floats): deg | agg | hA | hB  (~77 MB)
    float* ws  = (float*)d_ws;
    size_t NP  = ((size_t)N + 63) & ~(size_t)63;
    float* deg = ws;
    float* agg = ws + NP;
    float* hA  = agg + NP * 128;
    float* hB  = hA + NP * 128;

    const int BLK = 256;               // 8 wave32 waves per block

    // --- degrees ---
    fill0_f32<<<cdiv(N, BLK), BLK, 0, stream>>>(deg, N);
    deg_count<<<cdiv(E, BLK), BLK, 0, stream>>>(dst, deg, E);
    deg_invert<<<cdiv(N, BLK), BLK, 0, stream>>>(deg, N);

    const int aggN4   = N * 32;        // N*128 floats as float4
    const int scatThr = E * 32;        // 32 threads per edge
    const int slabs   = cdiv(N, 128);  // GEMM blocks (128-row slabs)

    // --- layer 0: x(128) -> hA(128), ReLU ---
    fill0_f32x4<<<cdiv(aggN4, BLK), BLK, 0, stream>>>((float4*)agg, aggN4);
    scatter_add_128<<<cdiv(scatThr, BLK), BLK, 0, stream>>>(x, src, dst, agg, E);
    sage_combine_wmma<<<slabs, BLK, 128 * HID * 2 * sizeof(float), stream>>>(
        x, agg, deg, Ws0, Wn0, b0, hA, N, 128, HID, 1);

    // --- layer 1: hA(128) -> hB(128), ReLU ---
    fill0_f32x4<<<cdiv(aggN4, BLK), BLK, 0, stream>>>((float4*)agg, aggN4);
    scatter_add_128<<<cdiv(scatThr, BLK), BLK, 0, stream>>>(hA, src, dst, agg, E);
    sage_combine_wmma<<<slabs, BLK, HID * HID * 2 * sizeof(float), stream>>>(
        hA, agg, deg, Ws1, Wn1, b1, hB, N, HID, HID, 1);

    // --- layer 2: hB(128) -> out(64), no activation ---
    fill0_f32x4<<<cdiv(aggN4, BLK), BLK, 0, stream>>>((float4*)agg, aggN4);
    scatter_add_128<<<cdiv(scatThr, BLK), BLK, 0, stream>>>(hB, src, dst, agg, E);
    sage_combine_wmma<<<slabs, BLK, HID * OUT * 2 * sizeof(float), stream>>>(
        hB, agg, deg, Ws2, Wn2, b2, out, N, HID, OUT, 0);
}